// spatialAttention_33758442946564
// MI455X (gfx1250) — compile-verified
//
#include <hip/hip_runtime.h>

#define NB 512
#define P  128
#define D  64
#define EPSF 1e-5f

#define LW 129   // padded stride for 128-col LDS rows (odd -> conflict-free strided reads)
#define LH 65    // padded stride for 64-col hidden rows

typedef __attribute__((ext_vector_type(2))) float v2f;
typedef __attribute__((ext_vector_type(8))) float v8f;

// One block: batch b, rows [row0, row0+16). 128 threads = 4 waves; each wave
// produces one 16x16 output tile (D = 4 tiles of 16 cols).
__global__ __launch_bounds__(128)
void spatial_attn_fused_kernel(const float* __restrict__ hid,    // [B,P,D]
                               const float* __restrict__ dist,   // [B,P,P]
                               const float* __restrict__ bear,   // [B,P,P]
                               const float* __restrict__ head,   // [B,P,P]
                               const float* __restrict__ smask,  // [B,P]
                               const float* __restrict__ dom,    // [12,12]
                               const float* __restrict__ Wg,     // [D,2D]
                               const float* __restrict__ bias,   // [D]
                               float* __restrict__ out)          // [B,P,D]
{
    __shared__ float lds_hid[P * LH];     // hidden[b]: [k][n], padded
    __shared__ float lds_W[D * LW];       // W as given: [n][k], padded
    __shared__ float lds_w[16 * LW];      // weights tile rows, padded
    __shared__ float lds_f[16 * LW];      // fused A = [wh | hidden rows]
    __shared__ float lds_dom[144];
    __shared__ float lds_mask[P];
    __shared__ float lds_bias[D];
    __shared__ float lds_inv[16];

    const int tid  = threadIdx.x;
    const int blk  = blockIdx.x;
    const int b_   = blk >> 3;
    const int row0 = (blk & 7) << 4;

    // ---- cooperative loads into LDS (coalesced global reads) ----
    const float* hb = hid + (size_t)b_ * P * D;
    for (int idx = tid; idx < P * D; idx += 128) {
        int k = idx >> 6, n = idx & 63;
        lds_hid[k * LH + n] = hb[idx];
    }
    for (int idx = tid; idx < D * 2 * D; idx += 128) {
        int n = idx >> 7, k = idx & 127;
        lds_W[n * LW + k] = Wg[idx];
    }
    for (int idx = tid; idx < 144; idx += 128) lds_dom[idx] = dom[idx];
    lds_mask[tid] = smask[b_ * P + tid];
    if (tid < D) lds_bias[tid] = bias[tid];
    __syncthreads();

    // ---- weights tile: col = tid, loop 16 rows (fully coalesced streams) ----
    {
        const int c = tid;
        const float mc = lds_mask[c];
        for (int r = 0; r < 16; ++r) {
            const int row = row0 + r;
            const size_t off = ((size_t)b_ * P + row) * P + c;
            float h  = head[off];
            float g  = bear[off];
            float dd = dist[off];
            int i1 = (int)floorf(h * (1.0f / 30.0f));
            int i2 = (int)floorf(g * (1.0f / 30.0f));
            i1 = min(max(i1, 0), 11);
            i2 = min(max(i2, 0), 11);
            float w = lds_dom[i1 * 12 + i2] - dd;
            w = fmaxf(w, 0.0f);
            if (c == row) w = 0.0f;
            if (lds_mask[row] * mc == 0.0f) w = 0.0f;
            float e = (w != 0.0f) ? __expf(w) : 0.0f;
            lds_w[r * LW + c] = e + EPSF;     // masked_exps + EPS
        }
    }
    __syncthreads();
    if (tid < 16) {                            // row sums -> 1/(sum+EPS)
        float s = 0.0f;
        for (int c = 0; c < P; ++c) s += lds_w[tid * LW + c];
        lds_inv[tid] = 1.0f / (s + EPSF);
    }
    __syncthreads();
    for (int r = 0; r < 16; ++r) lds_w[r * LW + tid] *= lds_inv[r];
    // hidden half of fused A while we are here
    for (int idx = tid; idx < 16 * D; idx += 128) {
        int m = idx >> 6, k = idx & 63;
        lds_f[m * LW + D + k] = lds_hid[(row0 + m) * LH + k];
    }
    __syncthreads();

    // ---- WMMA fragment coordinates (16x16x4 f32 layout) ----
    const int lane = tid & 31;
    const int wave = tid >> 5;
    const int n0   = wave << 4;               // this wave's 16-col tile
    const int m    = lane & 15;               // A row
    const int ks   = (lane >> 4) << 1;        // K sub-select: 0 or 2
    const int nn   = n0 + (lane & 15);        // B/D column

    // ---- stage B: wh[16x64] = weights[16x128] x hidden[128x64] ----
    v8f acc = {0.f, 0.f, 0.f, 0.f, 0.f, 0.f, 0.f, 0.f};
    for (int kb = 0; kb < P; kb += 4) {
        v2f a, bf;
        a.x  = lds_w[m * LW + kb + ks];
        a.y  = lds_w[m * LW + kb + ks + 1];
        bf.x = lds_hid[(kb + ks) * LH + nn];
        bf.y = lds_hid[(kb + ks + 1) * LH + nn];
        acc = __builtin_amdgcn_wmma_f32_16x16x4_f32(false, a, false, bf,
                                                    (short)0, acc, false, false);
    }
#pragma unroll
    for (int r = 0; r < 8; ++r) {             // D tile -> fused cols 0..63
        int mr = r + ((lane >> 4) << 3);
        lds_f[mr * LW + nn] = acc[r];
    }
    __syncthreads();

    // ---- stage C: out[16x64] = fused[16x128] x W^T[128x64] ----
    v8f acc2 = {0.f, 0.f, 0.f, 0.f, 0.f, 0.f, 0.f, 0.f};
    for (int kb = 0; kb < 2 * D; kb += 4) {
        v2f a, bf;
        a.x  = lds_f[m * LW + kb + ks];
        a.y  = lds_f[m * LW + kb + ks + 1];
        bf.x = lds_W[nn * LW + kb + ks];       // B[k][n] = W[n][k]
        bf.y = lds_W[nn * LW + kb + ks + 1];
        acc2 = __builtin_amdgcn_wmma_f32_16x16x4_f32(false, a, false, bf,
                                                     (short)0, acc2, false, false);
    }
#pragma unroll
    for (int r = 0; r < 8; ++r) {
        int mr = r + ((lane >> 4) << 3);
        float v = tanhf(acc2[r] + lds_bias[nn]);
        out[(((size_t)b_ * P) + row0 + mr) * D + nn] = v;
    }
}

extern "C" void kernel_launch(void* const* d_in, const int* in_sizes, int n_in,
                              void* d_out, int out_size, void* d_ws, size_t ws_size,
                              hipStream_t stream) {
    (void)in_sizes; (void)n_in; (void)out_size; (void)d_ws; (void)ws_size;
    const float* hid   = (const float*)d_in[0];
    const float* dist  = (const float*)d_in[1];
    const float* bear  = (const float*)d_in[2];
    const float* head  = (const float*)d_in[3];
    const float* smask = (const float*)d_in[4];
    const float* dom   = (const float*)d_in[5];
    const float* Wg    = (const float*)d_in[6];
    const float* bias  = (const float*)d_in[7];
    float* out = (float*)d_out;

    dim3 grid(NB * (P / 16));   // 512 * 8 = 4096 blocks
    dim3 block(128);            // 4 waves
    spatial_attn_fused_kernel<<<grid, block, 0, stream>>>(
        hid, dist, bear, head, smask, dom, Wg, bias, out);
}